// MultiSpark_19997367730506
// MI455X (gfx1250) — compile-verified
//
#include <hip/hip_runtime.h>
#include <math.h>

// ---------------------------------------------------------------------------
// MultiSpark on MI455X (gfx1250)
//   Phase 1: y = W @ (s*0.95)  via V_WMMA_F32_16X16X4_F32, K-split x8
//   Phase 1b: s_ws = sigmoid(sum_partials + noise)
//   Phase 2: force young sparks; sequential 256-step scan in ONE workgroup:
//            row read (+delta patch) -> relu+eps -> block scan -> searchsorted
//            -> Hebbian delta recorded -> s/energy/age updates
//   Phase 3: W_out = clip(W*0.999, +-2)  (streaming, float4)
//   Phase 4: apply the 256 Hebbian deltas (in order) into W_out
//   Phase 5: copy final s to output
// ---------------------------------------------------------------------------

typedef __attribute__((ext_vector_type(2))) float v2f;
typedef __attribute__((ext_vector_type(8))) float v8f;

#define KSPLIT          8
#define WAVES_PER_BLK   8
#define SCAN_THREADS    1024

__device__ __forceinline__ float sigmoidf(float x) {
    return 1.0f / (1.0f + __expf(-x));
}

// ---------------- Phase 1: WMMA matvec partials ----------------------------
// grid.x = (n/16/WAVES_PER_BLK) * KSPLIT, block = 256 threads (8 waves, wave32)
// partial[kchunk][row] = sum over k in chunk of W[row,k]*s[k]*0.95
__global__ void k_matvec_wmma(const float* __restrict__ W,
                              const float* __restrict__ s,
                              float* __restrict__ partial,
                              int n) {
    extern __shared__ float lds_s[];                 // n/KSPLIT floats
    const int chunk   = n / KSPLIT;
    const int kchunk  = blockIdx.x % KSPLIT;
    const int bandgrp = blockIdx.x / KSPLIT;
    const int k0      = kchunk * chunk;

    for (int i = threadIdx.x; i < chunk; i += blockDim.x)
        lds_s[i] = s[k0 + i] * 0.95f;                // STATE_DECAY folded in
    __syncthreads();

    const int wave = threadIdx.x >> 5;               // wave32
    const int lane = threadIdx.x & 31;
    const int m    = lane & 15;                      // row within 16-band
    const int half = lane >> 4;                      // K-pair select
    const int band = bandgrp * WAVES_PER_BLK + wave; // 16-row band index

    const float* rowp = W + (size_t)(band * 16 + m) * (size_t)n + k0;

    v8f c = {};                                      // f32 accumulator (column 0 used)
    #pragma unroll 4
    for (int kk = 0; kk < chunk; kk += 4) {
        // A tile 16x4: lane m<16 holds K=kk..kk+1, lane m+16 holds K=kk+2..kk+3
        v2f a = *(const v2f*)(rowp + kk + half * 2);
        // B tile 4x16: only column N==0 carries s; other columns zero
        float s0 = lds_s[kk + half * 2];
        float s1 = lds_s[kk + half * 2 + 1];
        v2f b;
        b.x = (m == 0) ? s0 : 0.0f;
        b.y = (m == 0) ? s1 : 0.0f;
        c = __builtin_amdgcn_wmma_f32_16x16x4_f32(
                false, a, false, b, (short)0, c, false, false);
    }

    // D column 0: lane0 -> rows band*16+0..7 (VGPR0..7), lane16 -> +8..15
    if (m == 0) {
        float* p = partial + (size_t)kchunk * n + band * 16 + half * 8;
        #pragma unroll
        for (int i = 0; i < 8; ++i) p[i] = c[i];
    }
}

// ---------------- Phase 1b: reduce partials + noise + sigmoid --------------
__global__ void k_reduce_sigmoid(const float* __restrict__ partial,
                                 const float* __restrict__ noise,
                                 float* __restrict__ s_ws, int n) {
    int i = blockIdx.x * blockDim.x + threadIdx.x;
    if (i < n) {
        float y = 0.0f;
        #pragma unroll
        for (int kc = 0; kc < KSPLIT; ++kc) y += partial[(size_t)kc * n + i];
        s_ws[i] = sigmoidf(y + noise[i]);
    }
}

// ---------------- Phase 2: forcing + sequential spark scan -----------------
// One workgroup of 1024 threads. Deltas (row,col,val) recorded in order.
__global__ void __launch_bounds__(SCAN_THREADS, 1)
k_scan(const float* __restrict__ W,
       float* __restrict__ s_ws,
       const int*   __restrict__ spark_pos,
       const float* __restrict__ spark_energy,
       const int*   __restrict__ spark_age,
       const float* __restrict__ u,
       int* __restrict__ delta_row, int* __restrict__ delta_col,
       float* __restrict__ delta_val,
       float* __restrict__ out_pos, float* __restrict__ out_energy,
       float* __restrict__ out_age,
       int n, int k) {
    extern __shared__ float lds[];
    float* row   = lds;                  // n floats (raw row, patched)
    float* scanA = lds + n;              // SCAN_THREADS floats
    float* scanB = scanA + SCAN_THREADS; // SCAN_THREADS floats
    __shared__ int   s_idx;
    __shared__ float s_target;

    const int t   = threadIdx.x;
    const int epc = n / SCAN_THREADS;    // elements per thread (16)

    // Force young sparks to 1.0 (serial: deterministic duplicate handling)
    if (t == 0) {
        for (int i = 0; i < k; ++i)
            if (spark_age[i] < 5) s_ws[spark_pos[i]] = 1.0f;
    }
    __syncthreads();

    for (int i = 0; i < k; ++i) {
        const int prev = spark_pos[i];
        const float* wrow = W + (size_t)prev * (size_t)n;

        // stream raw row into LDS (float4), prefetch next spark's row
        for (int j = t; j < n / 4; j += SCAN_THREADS)
            ((float4*)row)[j] = ((const float4*)wrow)[j];
        if (i + 1 < k) {
            const float* nr = W + (size_t)spark_pos[i + 1] * (size_t)n;
            __builtin_prefetch(nr + t * epc, 0, 1);  // global_prefetch_b8
        }
        __syncthreads();

        // apply earlier Hebbian deltas hitting this row (serial, in order)
        if (t == 0) {
            for (int j = 0; j < i; ++j)
                if (delta_row[j] == prev) row[delta_col[j]] = delta_val[j];
        }
        __syncthreads();

        // weights = relu(row)+1e-6; per-thread contiguous chunk sums
        float wloc[16];
        float local = 0.0f;
        #pragma unroll
        for (int e = 0; e < 16; ++e) {
            float v = row[t * epc + e];
            v = fmaxf(v, 0.0f) + 1e-6f;
            wloc[e] = v;
            local += v;
        }
        scanA[t] = local;
        __syncthreads();

        // Hillis-Steele inclusive scan over the 1024 chunk sums
        float* a = scanA; float* b = scanB;
        for (int off = 1; off < SCAN_THREADS; off <<= 1) {
            float v = a[t];
            if (t >= off) v += a[t - off];
            b[t] = v;
            __syncthreads();
            float* tmp = a; a = b; b = tmp;
        }

        if (t == 0) {
            s_target = u[i] * a[SCAN_THREADS - 1];   // u * c[-1]
            s_idx    = n - 1;                        // clamp fallback
        }
        __syncthreads();

        // searchsorted: first index with inclusive cumsum >= target
        const float target = s_target;
        const float incl = a[t];
        const float excl = (t == 0) ? 0.0f : a[t - 1];
        if (incl >= target && excl < target) {       // unique owner thread
            float run = excl;
            int idx = t * epc + epc - 1;
            #pragma unroll
            for (int e = 0; e < 16; ++e) {
                run += wloc[e];
                if (run >= target) { idx = t * epc + e; break; }
            }
            s_idx = idx;
        }
        __syncthreads();
        const int nxt = s_idx;

        if (t == 0) {
            // Hebbian edge update at (nxt, prev), honoring earlier deltas
            float base = W[(size_t)nxt * (size_t)n + prev];
            for (int j = 0; j < i; ++j)
                if (delta_row[j] == nxt && delta_col[j] == prev)
                    base = delta_val[j];
            float nv = base * (1.0f - 0.05f) + s_ws[prev] * 0.05f;
            delta_row[i] = nxt; delta_col[i] = prev; delta_val[i] = nv;

            float e = spark_energy[i] * 0.98f;       // SPARK_ENERGY_DECAY
            s_ws[nxt] = e;                           // always set
            int  age   = spark_age[i] + 1;
            bool reset = e < 0.05f;                  // SPARK_MIN_ENERGY
            out_pos[i]    = (float)(reset ? (i % n) : nxt);
            out_energy[i] = reset ? 1.0f : e;
            out_age[i]    = (float)(reset ? 0 : age);
        }
        __syncthreads();
    }
}

// ---------------- Phase 3: W_out = clip(W*0.999, +-2) ----------------------
__global__ void k_decay_clip(const float4* __restrict__ in,
                             float4* __restrict__ out, size_t n4) {
    size_t i = (size_t)blockIdx.x * blockDim.x + threadIdx.x;
    size_t stride = (size_t)gridDim.x * blockDim.x;
    for (; i < n4; i += stride) {
        float4 v = in[i];
        v.x = fminf(fmaxf(v.x * 0.999f, -2.0f), 2.0f);
        v.y = fminf(fmaxf(v.y * 0.999f, -2.0f), 2.0f);
        v.z = fminf(fmaxf(v.z * 0.999f, -2.0f), 2.0f);
        v.w = fminf(fmaxf(v.w * 0.999f, -2.0f), 2.0f);
        out[i] = v;
    }
}

// ---------------- Phase 4: apply Hebbian deltas into W_out (in order) ------
__global__ void k_fixup(float* __restrict__ Wout,
                        const int* __restrict__ dr, const int* __restrict__ dc,
                        const float* __restrict__ dv, int k, int n) {
    if (blockIdx.x == 0 && threadIdx.x == 0) {
        for (int j = 0; j < k; ++j) {
            float v = fminf(fmaxf(dv[j] * 0.999f, -2.0f), 2.0f);
            Wout[(size_t)dr[j] * (size_t)n + dc[j]] = v;
        }
    }
}

// ---------------- Phase 5: copy final s ------------------------------------
__global__ void k_copy_s(const float* __restrict__ s_ws,
                         float* __restrict__ out, int n) {
    int i = blockIdx.x * blockDim.x + threadIdx.x;
    if (i < n) out[i] = s_ws[i];
}

// ---------------------------------------------------------------------------
extern "C" void kernel_launch(void* const* d_in, const int* in_sizes, int n_in,
                              void* d_out, int out_size, void* d_ws, size_t ws_size,
                              hipStream_t stream) {
    const float* W        = (const float*)d_in[0];
    const float* s        = (const float*)d_in[1];
    const float* noise    = (const float*)d_in[2];
    const float* u        = (const float*)d_in[3];
    const int*   spark_pos    = (const int*)d_in[4];
    const float* spark_energy = (const float*)d_in[5];
    const int*   spark_age    = (const int*)d_in[6];

    const int n = in_sizes[1];   // 16384
    const int k = in_sizes[3];   // 256

    // workspace layout
    char* ws = (char*)d_ws;
    float* partial = (float*)ws;                      ws += (size_t)KSPLIT * n * sizeof(float);
    float* s_ws    = (float*)ws;                      ws += (size_t)n * sizeof(float);
    int*   drow    = (int*)ws;                        ws += (size_t)k * sizeof(int);
    int*   dcol    = (int*)ws;                        ws += (size_t)k * sizeof(int);
    float* dval    = (float*)ws;                      ws += (size_t)k * sizeof(float);

    // output layout: W [n*n] | s [n] | pos [k] | energy [k] | age [k]
    float* outW   = (float*)d_out;
    float* outS   = outW + (size_t)n * n;
    float* outPos = outS + n;
    float* outE   = outPos + k;
    float* outAge = outE + k;

    // Phase 1: WMMA matvec partials
    {
        int blocks = (n / 16 / WAVES_PER_BLK) * KSPLIT;
        size_t shmem = (size_t)(n / KSPLIT) * sizeof(float);
        k_matvec_wmma<<<blocks, 32 * WAVES_PER_BLK, shmem, stream>>>(W, s, partial, n);
    }
    // Phase 1b
    k_reduce_sigmoid<<<(n + 255) / 256, 256, 0, stream>>>(partial, noise, s_ws, n);

    // Phase 2: scan (single workgroup)
    {
        size_t shmem = (size_t)n * sizeof(float) + 2 * SCAN_THREADS * sizeof(float);
        k_scan<<<1, SCAN_THREADS, shmem, stream>>>(W, s_ws, spark_pos, spark_energy,
                                                   spark_age, u, drow, dcol, dval,
                                                   outPos, outE, outAge, n, k);
    }

    // Phase 3: streaming decay+clip of W into output
    {
        size_t n4 = (size_t)n * n / 4;
        k_decay_clip<<<8192, 256, 0, stream>>>((const float4*)W, (float4*)outW, n4);
    }
    // Phase 4: patch the 256 updated edges
    k_fixup<<<1, 32, 0, stream>>>(outW, drow, dcol, dval, k, n);
    // Phase 5: final s
    k_copy_s<<<(n + 255) / 256, 256, 0, stream>>>(s_ws, outS, n);
}